// LocalNetwork_3393024163963
// MI455X (gfx1250) — compile-verified
//
#include <hip/hip_runtime.h>
#include <stdint.h>

// Problem constants (from reference)
constexpr int Bn = 256;
constexpr int Dn = 15, Hn = 64, Wn = 128;
constexpr int dn = 5,  hn = 16, wn = 32;
constexpr int LDEP = (dn + 2) * hn * wn;   // 3584
constexpr int LLON = dn * hn * (wn + 2);   // 2720
constexpr int LLAT = dn * (hn + 2) * wn;   // 2880
constexpr int CO   = dn * hn * wn;         // 2560 coarse elems / batch
constexpr int DHW  = Dn * Hn * Wn;         // 122880

typedef float v4f __attribute__((ext_vector_type(4)));   // native vector for NT stores

// ---------------------------------------------------------------------------
// Kernel 1: 3x4x4 average pool. One block per (b, di, hi-group-of-4) tile.
// The tile's 48 rows (128 floats each) are staged into LDS with CDNA5 async
// global->LDS B128 copies (ASYNCcnt path). Input is streamed exactly once, so
// the async loads carry a non-temporal hint to keep L2 free for reused data.
// ---------------------------------------------------------------------------
__global__ __launch_bounds__(256) void pool_kernel(const float* __restrict__ x,
                                                   float* __restrict__ xs_g) {
    __shared__ float tile[48 * Wn];            // 24 KB
    const int blk  = blockIdx.x;               // b*20 + di*4 + g
    const int b    = blk / 20;
    const int t20  = blk % 20;
    const int di   = t20 >> 2;
    const int g    = t20 & 3;
    const int tid  = threadIdx.x;
    const int lane = tid & 31;
    const int wv   = tid >> 5;                 // 8 waves (wave32)

    const size_t   xbase = (size_t)b * DHW;
    const uint32_t lds0  = (uint32_t)(uintptr_t)(&tile[0]);

    // Each wave issues 6 async B128 loads: one input row per instruction
    // (32 lanes x 16B = 512B = one full row of 128 floats).
    #pragma unroll
    for (int k = 0; k < 6; ++k) {
        const int r   = wv + (k << 3);                 // 0..47, unique per wave
        const int ddx = 3 * di + (r >> 4);             // depth row
        const int hhx = (g << 4) + (r & 15);           // h row
        const uint64_t gsrc =
            (uint64_t)(uintptr_t)(x + xbase + (size_t)(ddx * Hn + hhx) * Wn + lane * 4);
        const uint32_t ldst = lds0 + (uint32_t)((r * Wn + lane * 4) * 4);
        asm volatile("global_load_async_to_lds_b128 %0, %1, off th:TH_LOAD_NT"
                     :: "v"(ldst), "v"(gsrc) : "memory");
    }
    asm volatile("s_wait_asynccnt 0x0" ::: "memory");
    __syncthreads();

    // 128 coarse outputs per tile: thread t -> (hi_local = t/32, wi = t%32)
    if (tid < 128) {
        const int hi_l = tid >> 5;
        const int wi   = tid & 31;
        float s = 0.0f;
        #pragma unroll
        for (int dl = 0; dl < 3; ++dl) {
            #pragma unroll
            for (int j = 0; j < 4; ++j) {
                const v4f q =
                    *(const v4f*)&tile[(dl * 16 + hi_l * 4 + j) * Wn + wi * 4];
                s += (q.x + q.y) + (q.z + q.w);
            }
        }
        const int hi = (g << 2) + hi_l;
        xs_g[(size_t)b * CO + (di * hn + hi) * wn + wi] = s * (1.0f / 48.0f);
    }
}

// ---------------------------------------------------------------------------
// Kernel 2: the three locally-connected width-3 conv+ReLU stages, entirely in
// LDS, one block per batch element. Padded flat buffers reproduce the
// reference's pad/transpose/slice chain exactly:
//   stage A (depth): flat l = (wi*16+hi)*7 + d7,   keep d7 in [1,5]
//   stage B (lon):   flat l = (di*16+hi)*34 + s,   keep s  in [1,32]
//   stage C (lat):   flat l = (di*32+wi)*18 + s,   keep s  in [1,16]
// Buffer overlays: sPA (3586) doubles as pC (2882); sXS (2560) doubles as v.
// ---------------------------------------------------------------------------
__global__ __launch_bounds__(256) void conv_kernel(
    const float* __restrict__ xs_g,
    const float* __restrict__ w_dep, const float* __restrict__ b_dep,
    const float* __restrict__ w_lon, const float* __restrict__ b_lon,
    const float* __restrict__ w_lat, const float* __restrict__ b_lat,
    float* __restrict__ v_g) {
    __shared__ float sXS[CO];          // coarse input, later reused as v output
    __shared__ float sPA[LDEP + 2];    // padded depth-flat; reused as padded lat-flat
    __shared__ float sPB[LLON + 2];    // padded lon-flat
    const int b   = blockIdx.x;
    const int tid = threadIdx.x;

    for (int i = tid; i < CO; i += 256)        sXS[i] = xs_g[(size_t)b * CO + i];
    for (int i = tid; i < LLON + 2; i += 256)  sPB[i] = 0.0f;
    if (tid == 0) { sPA[0] = 0.0f; sPA[LDEP + 1] = 0.0f; }
    __syncthreads();

    // build padded depth-flat input
    for (int l = tid; l < LDEP; l += 256) {
        const int d7 = l % 7;
        const int hi = (l / 7) & 15;
        const int wi = l / 112;
        sPA[1 + l] = (d7 >= 1 && d7 <= 5) ? sXS[((d7 - 1) * hn + hi) * wn + wi] : 0.0f;
    }
    __syncthreads();

    // stage A: depth conv -> scatter kept positions into padded lon-flat
    for (int l = tid; l < LDEP; l += 256) {
        float v = w_dep[3 * l] * sPA[l] + w_dep[3 * l + 1] * sPA[l + 1]
                + w_dep[3 * l + 2] * sPA[l + 2] + b_dep[l];
        v = fmaxf(v, 0.0f);
        const int d7 = l % 7;
        if (d7 >= 1 && d7 <= 5) {
            const int hi = (l / 7) & 15;
            const int wi = l / 112;
            sPB[1 + ((d7 - 1) * hn + hi) * 34 + (wi + 1)] = v;
        }
    }
    __syncthreads();

    // zero padded lat-flat (overlays sPA, now dead)
    for (int i = tid; i < LLAT + 2; i += 256) sPA[i] = 0.0f;
    __syncthreads();

    // stage B: lon conv -> scatter kept positions into padded lat-flat
    for (int l = tid; l < LLON; l += 256) {
        float v = w_lon[3 * l] * sPB[l] + w_lon[3 * l + 1] * sPB[l + 1]
                + w_lon[3 * l + 2] * sPB[l + 2] + b_lon[l];
        v = fmaxf(v, 0.0f);
        const int s = l % 34;
        if (s >= 1 && s <= 32) {
            const int hi = (l / 34) & 15;
            const int di = l / 544;
            sPA[1 + (di * wn + (s - 1)) * 18 + (hi + 1)] = v;
        }
    }
    __syncthreads();

    // stage C: lat conv -> kept positions into sXS as v[di][hi][wi]
    for (int l = tid; l < LLAT; l += 256) {
        float v = w_lat[3 * l] * sPA[l] + w_lat[3 * l + 1] * sPA[l + 1]
                + w_lat[3 * l + 2] * sPA[l + 2] + b_lat[l];
        v = fmaxf(v, 0.0f);
        const int s = l % 18;
        if (s >= 1 && s <= 16) {
            const int wi = (l / 18) & 31;
            const int di = l / 576;
            sXS[(di * hn + (s - 1)) * wn + wi] = v;
        }
    }
    __syncthreads();

    for (int i = tid; i < CO; i += 256) v_g[(size_t)b * CO + i] = sXS[i];
}

// ---------------------------------------------------------------------------
// Kernel 3: nearest-neighbor broadcast back to (B,15,64,128). One block per
// (b, dd) slab; each wave stores full 512B rows with non-temporal 128-bit
// stores (output is written once, never re-read by the GPU).
// ---------------------------------------------------------------------------
__global__ __launch_bounds__(256) void bcast_kernel(const float* __restrict__ v_g,
                                                    float* __restrict__ out) {
    const int blk  = blockIdx.x;          // b*15 + dd
    const int b    = blk / Dn;
    const int dd   = blk % Dn;
    const int di   = dd / 3;
    const int tid  = threadIdx.x;
    const int lane = tid & 31;
    const int wv   = tid >> 5;

    const size_t obase = (size_t)b * DHW + (size_t)dd * (Hn * Wn);
    const float* vb    = v_g + (size_t)b * CO + di * hn * wn;

    #pragma unroll
    for (int k = 0; k < 8; ++k) {
        const int   hh  = wv + (k << 3);            // 0..63
        const float val = vb[(hh >> 2) * wn + lane];
        const v4f   q   = {val, val, val, val};
        __builtin_nontemporal_store(
            q, reinterpret_cast<v4f*>(out + obase + (size_t)hh * Wn + lane * 4));
    }
}

extern "C" void kernel_launch(void* const* d_in, const int* in_sizes, int n_in,
                              void* d_out, int out_size, void* d_ws, size_t ws_size,
                              hipStream_t stream) {
    const float* x     = (const float*)d_in[0];
    const float* w_dep = (const float*)d_in[1];
    const float* b_dep = (const float*)d_in[2];
    const float* w_lon = (const float*)d_in[3];
    const float* b_lon = (const float*)d_in[4];
    const float* w_lat = (const float*)d_in[5];
    const float* b_lat = (const float*)d_in[6];
    float* out = (float*)d_out;

    float* xs_ws = (float*)d_ws;                    // B*2560 floats
    float* v_ws  = xs_ws + (size_t)Bn * CO;         // B*2560 floats (5.25 MB total)

    pool_kernel<<<Bn * 20, 256, 0, stream>>>(x, xs_ws);
    conv_kernel<<<Bn, 256, 0, stream>>>(xs_ws, w_dep, b_dep, w_lon, b_lon,
                                        w_lat, b_lat, v_ws);
    bcast_kernel<<<Bn * Dn, 256, 0, stream>>>(v_ws, out);
}